// Gauss2D_54047868453283
// MI455X (gfx1250) — compile-verified
//
#include <hip/hip_runtime.h>

typedef __attribute__((ext_vector_type(2))) float v2f;
typedef __attribute__((ext_vector_type(4))) float v4f;
typedef __attribute__((ext_vector_type(8))) float v8f;

#define HDIM 15
#define TSTRIDE 272   // floats between the two per-wave LDS tiles (16B aligned, +16 bank rotation)

__device__ __forceinline__ v8f wmma_k4(v2f a, v2f b, v8f c) {
    // D = A(16x4,f32) * B(4x16,f32) + C(16x16,f32)
    return __builtin_amdgcn_wmma_f32_16x16x4_f32(
        /*neg_a=*/false, a, /*neg_b=*/false, b,
        /*c_mod=*/(short)0, c, /*reuse_a=*/false, /*reuse_b=*/false);
}

// Fast sigmoid: v_exp_f32 + v_rcp_f32 (TRANS ops co-execute with WMMA).
__device__ __forceinline__ float sigmoidf_fast(float v) {
    return __builtin_amdgcn_rcpf(1.0f + __expf(-v));
}

// Store C/D-layout activations into LDS as T[feature*16 + point].
// C/D layout: VGPR r holds M=r (lanes 0-15) / M=r+8 (lanes 16-31), N = lane&15.
__device__ __forceinline__ void store_act(v8f c, float* T, int nc, int half) {
    float* dst = T + nc * 16 + half * 8;
    v4f lo; lo.x = c[0]; lo.y = c[1]; lo.z = c[2]; lo.w = c[3];
    v4f hi; hi.x = c[4]; hi.y = c[5]; hi.z = c[6]; hi.w = c[7];
    *(v4f*)dst       = lo;
    *(v4f*)(dst + 4) = hi;
}

// A chunk j (16x4): VGPR0 = K=4j (lanes0-15) | 4j+2 (lanes16-31); VGPR1 = +1.
__device__ __forceinline__ v2f load_a_chunk(const float* T, int j, int nc, int half) {
    const int k = 4 * j + 2 * half;
    v2f a;
    a.x = T[(k    ) * 16 + nc];
    a.y = T[(k + 1) * 16 + nc];
    return a;
}

// Two independent tiles through one hidden layer: interleaved WMMA chains.
__device__ __forceinline__ void hidden_layer2(v8f& c0, v8f& c1, const v2f* Bc,
                                              float bias, float* T,
                                              int nc, int half) {
    store_act(c0, T, nc, half);
    store_act(c1, T + TSTRIDE, nc, half);
    asm volatile("s_wait_dscnt 0" ::: "memory");

    v8f a0 = {}, a1 = {};
#pragma unroll
    for (int j = 0; j < 4; ++j) {
        v2f x0 = load_a_chunk(T, j, nc, half);
        v2f x1 = load_a_chunk(T + TSTRIDE, j, nc, half);
        a0 = wmma_k4(x0, Bc[j], a0);
        a1 = wmma_k4(x1, Bc[j], a1);
    }
#pragma unroll
    for (int i = 0; i < 8; ++i) {
        float v0 = a0[i] + bias;
        float v1 = a1[i] + bias;
        a0[i] = v0 * sigmoidf_fast(v0);   // SiLU
        a1[i] = v1 * sigmoidf_fast(v1);
    }
    c0 = a0;
    c1 = a1;
}

__global__ __launch_bounds__(256) void mlp_wmma_kernel(
    const float* __restrict__ x,
    const float* __restrict__ W1, const float* __restrict__ b1,
    const float* __restrict__ W2, const float* __restrict__ b2,
    const float* __restrict__ W3, const float* __restrict__ b3,
    const float* __restrict__ W4, const float* __restrict__ b4,
    float* __restrict__ out, int numPoints)
{
    __shared__ float lds[8][2 * TSTRIDE];   // two transpose tiles per wave

    const int lane     = threadIdx.x & 31;
    const int wib      = threadIdx.x >> 5;
    const int waveId   = blockIdx.x * (blockDim.x >> 5) + wib;
    const int numWaves = gridDim.x * (blockDim.x >> 5);
    float* T = lds[wib];

    const int nc   = lane & 15;   // N (column/feature) index; also M for A loads
    const int half = lane >> 4;   // 0: lanes 0-15, 1: lanes 16-31

    // ---- Stage B matrices in registers (zero-padded to K=16 / N=16) ----
    // B chunk layout (4x16): VGPR0 = row K0 (lanes0-15) | K2 (lanes16-31); VGPR1 = K1 | K3
    const int k0 = 2 * half;
    const int k1 = 2 * half + 1;
    v2f B1;
    B1.x = (k0 < 2 && nc < HDIM) ? W1[k0 * HDIM + nc] : 0.0f;
    B1.y = (k1 < 2 && nc < HDIM) ? W1[k1 * HDIM + nc] : 0.0f;
    v2f B2c[4], B3c[4];
#pragma unroll
    for (int j = 0; j < 4; ++j) {
        const int ka = 4 * j + k0;
        const int kb = 4 * j + k1;
        B2c[j].x = (ka < HDIM && nc < HDIM) ? W2[ka * HDIM + nc] : 0.0f;
        B2c[j].y = (kb < HDIM && nc < HDIM) ? W2[kb * HDIM + nc] : 0.0f;
        B3c[j].x = (ka < HDIM && nc < HDIM) ? W3[ka * HDIM + nc] : 0.0f;
        B3c[j].y = (kb < HDIM && nc < HDIM) ? W3[kb * HDIM + nc] : 0.0f;
    }
    const float bias1 = (nc < HDIM) ? b1[nc] : 0.0f;
    const float bias2 = (nc < HDIM) ? b2[nc] : 0.0f;
    const float bias3 = (nc < HDIM) ? b3[nc] : 0.0f;

    // Layer-4 weights are wave-uniform -> SGPRs via scalar loads.
    float w4s[HDIM];
#pragma unroll
    for (int k = 0; k < HDIM; ++k) w4s[k] = W4[k];
    const float bias4 = b4[0];

    // All 32 lanes load one point each (fully coalesced 256B per wave).
    auto load_x_pair = [&](int sIdx) -> v2f {
        v2f a; a.x = 0.0f; a.y = 0.0f;
        const int row = sIdx * 32 + lane;
        if (row < numPoints) {
            a = *(const v2f*)(x + (size_t)row * 2);
        }
        return a;
    };

    const int numSuper = (numPoints + 31) >> 5;   // 32-point super-tiles

    // Software-pipelined x load for the first super-tile.
    v2f xv = load_x_pair(waveId);

    for (int s = waveId; s < numSuper; s += numWaves) {
        // Issue next super-tile's load now; consumed next iteration.
        v2f xvn = load_x_pair(s + numWaves);

        // Half-swap so lanes 0-15 also see points 16-31 (for tile 1's A matrix).
        v2f xs;
        xs.x = __int_as_float(__builtin_amdgcn_ds_swizzle(__float_as_int(xv.x), 0x401f));
        xs.y = __int_as_float(__builtin_amdgcn_ds_swizzle(__float_as_int(xv.y), 0x401f));

        // A (16x4): K=0,1 = x[m][0..1] in lanes 0-15; K=2,3 = 0 in lanes 16-31.
        v2f A0, A1;
        A0.x = (half == 0) ? xv.x : 0.0f;
        A0.y = (half == 0) ? xv.y : 0.0f;
        A1.x = (half == 0) ? xs.x : 0.0f;
        A1.y = (half == 0) ? xs.y : 0.0f;

        // ---- Layer 1: silu(x @ W1 + b1) — one K=4 WMMA per tile
        v8f c0 = {}, c1 = {};
        c0 = wmma_k4(A0, B1, c0);
        c1 = wmma_k4(A1, B1, c1);
#pragma unroll
        for (int i = 0; i < 8; ++i) {
            float v0 = c0[i] + bias1;
            float v1 = c1[i] + bias1;
            c0[i] = v0 * sigmoidf_fast(v0);
            c1[i] = v1 * sigmoidf_fast(v1);
        }

        // ---- Layers 2 & 3: silu(h @ W + b), 4 WMMAs per tile each
        hidden_layer2(c0, c1, B2c, bias2, T, nc, half);
        hidden_layer2(c0, c1, B3c, bias3, T, nc, half);

        // ---- Layer 4: per-lane dot product out of the LDS transpose.
        // h3 element (point m, feature k) sits at T[tile*TSTRIDE + k*16 + m].
        store_act(c0, T, nc, half);
        store_act(c1, T + TSTRIDE, nc, half);
        asm volatile("s_wait_dscnt 0" ::: "memory");

        const float* Tm = T + half * TSTRIDE + nc;  // lane -> its point's column
        float acc = bias4;
#pragma unroll
        for (int k = 0; k < HDIM; ++k) {
            acc = fmaf(Tm[k * 16], w4s[k], acc);
        }
        const float o = sigmoidf_fast(acc);

        const int row = s * 32 + lane;
        if (row < numPoints) {
            out[row] = o;    // fully coalesced: 32 lanes, 1 float each
        }
        asm volatile("s_wait_dscnt 0" ::: "memory"); // drain before T is reused

        xv = xvn;
    }
}

extern "C" void kernel_launch(void* const* d_in, const int* in_sizes, int n_in,
                              void* d_out, int out_size, void* d_ws, size_t ws_size,
                              hipStream_t stream) {
    const float* x  = (const float*)d_in[0];
    const float* W1 = (const float*)d_in[1];
    const float* b1 = (const float*)d_in[2];
    const float* W2 = (const float*)d_in[3];
    const float* b2 = (const float*)d_in[4];
    const float* W3 = (const float*)d_in[5];
    const float* b3 = (const float*)d_in[6];
    const float* W4 = (const float*)d_in[7];
    const float* b4 = (const float*)d_in[8];
    float* out = (float*)d_out;

    const int n = in_sizes[0] / 2;            // number of points
    const int numSuper = (n + 31) >> 5;       // 32-point super-tiles

    // 8 waves/block; ~4 super-tiles (128 points) per wave.
    int blocks = (numSuper + 31) / 32;
    if (blocks < 1) blocks = 1;
    if (blocks > 16384) blocks = 16384;

    mlp_wmma_kernel<<<blocks, 256, 0, stream>>>(
        x, W1, b1, W2, b2, W3, b3, W4, b4, out, n);
}